// PYG_DeepGCNEncoder_12120397710136
// MI455X (gfx1250) — compile-verified
//
#include <hip/hip_runtime.h>
#include <hip/hip_bf16.h>
#include <math.h>

#define NN 50000
#define EE 800000
#define DD 64
#define HH 128
#define LL 3
#define HD 8      // HDIM
#define ED 8      // EDIM
#define NT (NN/16)          // 3125 node tiles (exact)
#define EPS_MSG 1e-7f
#define BN_EPS  1e-5f
#define LN_EPS  1e-5f

typedef __attribute__((ext_vector_type(2))) float v2f;
typedef __attribute__((ext_vector_type(8))) float v8f;

// ---------------------------------------------------------------- utilities
__device__ __forceinline__ void atomicMaxF(float* addr, float val) {
    // IEEE ordered-int trick: works for mixed-sign contents, init = -inf
    if (val >= 0.0f) atomicMax((int*)addr, __float_as_int(val));
    else             atomicMin((unsigned int*)addr, __float_as_uint(val));
}

__device__ __forceinline__ float edge_feat(const float* __restrict__ ea,
                                           const float* __restrict__ eW,
                                           const float* __restrict__ eb,
                                           int e, int c) {
    // e[e][c] = edge_attr[e] . edge_W[:,c] + edge_b[c]  (recomputed, never stored)
    float acc = eb[c];
#pragma unroll
    for (int k = 0; k < ED; ++k) acc += ea[e * ED + k] * eW[k * DD + c];
    return acc;
}

// ---------------------------------------------------------------- encoders
__global__ void node_enc(const float* __restrict__ x, const float* __restrict__ W,
                         const float* __restrict__ b, float* __restrict__ h) {
    int gid = blockIdx.x * 256 + threadIdx.x;
    if (gid >= NN * DD) return;
    int n = gid >> 6, d = gid & 63;
    float acc = b[d];
#pragma unroll
    for (int k = 0; k < HD; ++k) acc += x[n * HD + k] * W[k * DD + d];
    h[gid] = acc;
}

// ---------------------------------------------------------------- per-layer
__global__ void layer_init(float* __restrict__ maxbuf, float* __restrict__ denom,
                           float* __restrict__ num, float* __restrict__ stats) {
    int gid = blockIdx.x * 256 + threadIdx.x;
    if (gid < 128) stats[gid] = 0.0f;
    if (gid < NN * DD) {
        maxbuf[gid] = -__builtin_inff();
        denom[gid]  = 0.0f;
        num[gid]    = 0.0f;
    }
}

__global__ void bn_stats(const float* __restrict__ h, float* __restrict__ stats) {
    int c = threadIdx.x & 63;
    int rl = threadIdx.x >> 6;              // 0..3
    float s = 0.0f, s2 = 0.0f;
    for (int r = blockIdx.x * 4 + rl; r < NN; r += gridDim.x * 4) {
        float v = h[r * DD + c];
        s += v; s2 += v * v;
    }
    __shared__ float sh[2][4][64];
    sh[0][rl][c] = s; sh[1][rl][c] = s2;
    __syncthreads();
    if (rl == 0) {
        s  = sh[0][0][c] + sh[0][1][c] + sh[0][2][c] + sh[0][3][c];
        s2 = sh[1][0][c] + sh[1][1][c] + sh[1][2][c] + sh[1][3][c];
        atomicAdd(&stats[c], s);
        atomicAdd(&stats[64 + c], s2);
    }
}

__global__ void bn_apply(const float* __restrict__ h, const float* __restrict__ stats,
                         const float* __restrict__ g, const float* __restrict__ beta,
                         float* __restrict__ hn) {
    int gid = blockIdx.x * 256 + threadIdx.x;
    if (gid >= NN * DD) return;
    int c = gid & 63;
    float mu  = stats[c] * (1.0f / NN);
    float var = stats[64 + c] * (1.0f / NN) - mu * mu;
    float v = (h[gid] - mu) * rsqrtf(var + BN_EPS) * g[c] + beta[c];
    hn[gid] = fmaxf(v, 0.0f);
}

__global__ void edge_max(const int* __restrict__ ei, const float* __restrict__ ea,
                         const float* __restrict__ eW, const float* __restrict__ eb,
                         const float* __restrict__ hn, const float* __restrict__ tptr,
                         float* __restrict__ maxbuf) {
    int gid = blockIdx.x * 256 + threadIdx.x;
    if (gid >= EE * DD) return;
    int e = gid >> 6, c = gid & 63;
    int src = ei[e], dst = ei[EE + e];
    float msg = fmaxf(hn[src * DD + c] + edge_feat(ea, eW, eb, e, c), 0.0f) + EPS_MSG;
    atomicMaxF(&maxbuf[dst * DD + c], msg * tptr[0]);
}

__global__ void edge_sum(const int* __restrict__ ei, const float* __restrict__ ea,
                         const float* __restrict__ eW, const float* __restrict__ eb,
                         const float* __restrict__ hn, const float* __restrict__ tptr,
                         const float* __restrict__ maxbuf,
                         float* __restrict__ denom, float* __restrict__ num) {
    int gid = blockIdx.x * 256 + threadIdx.x;
    if (gid >= EE * DD) return;
    int e = gid >> 6, c = gid & 63;
    int src = ei[e], dst = ei[EE + e];
    float msg = fmaxf(hn[src * DD + c] + edge_feat(ea, eW, eb, e, c), 0.0f) + EPS_MSG;
    float m = maxbuf[dst * DD + c];
    if (m == -__builtin_inff()) m = 0.0f;   // isfinite clamp (only -inf possible)
    float a = __expf(msg * tptr[0] - m);
    atomicAdd(&denom[dst * DD + c], a);
    atomicAdd(&num[dst * DD + c], a * msg);
}

// ------------------------------------------------- fused agg + MLP (WMMA f32)
// one wave32 per 16-node tile; 4 waves per block; fully fp32 via
// V_WMMA_F32_16X16X4_F32 (16 K-steps GEMM1, 32 K-steps GEMM2)
__global__ __launch_bounds__(128) void mlp_wmma(
        const float* __restrict__ hn, const float* __restrict__ num,
        const float* __restrict__ denom,
        const float* __restrict__ W1, const float* __restrict__ b1,
        const float* __restrict__ lg, const float* __restrict__ lb,
        const float* __restrict__ W2, const float* __restrict__ b2,
        float* __restrict__ h) {
    __shared__ float sOut[4][16][DD + 1];   // +1 pad: kill LDS bank conflicts
    __shared__ float sY[4][16][HH + 1];
    const int wave = threadIdx.x >> 5;
    const int lane = threadIdx.x & 31;
    const int tile = blockIdx.x * 4 + wave;
    if (tile >= NT) return;                 // wave-uniform: EXEC stays all-ones
    const int row0 = tile * 16;
    float (*out)[DD + 1] = sOut[wave];
    float (*y)[HH + 1]   = sY[wave];

    // stage out = softmax_agg + hn   (GENConv root residual)
    for (int idx = lane; idx < 16 * DD; idx += 32) {
        int r = idx >> 6, c = idx & 63;
        int g = (row0 + r) * DD + c;
        float agg = num[g] / fmaxf(denom[g], 1e-16f);
        out[r][c] = agg + hn[g];
    }

    const int rA   = lane & 15;             // A row / B column within tile
    const int kOff = (lane >> 4) << 1;      // K pair: lanes 0-15 -> {0,1}, 16-31 -> {2,3}
    const int mB   = (lane >> 4) << 3;      // C/D row base: 0 or 8

    // GEMM1: y = out(16x64) @ W1(64x128) + b1
    for (int ct = 0; ct < HH / 16; ++ct) {
        v8f acc = {};
        const int ncol = ct * 16 + rA;
#pragma unroll
        for (int ks = 0; ks < DD / 4; ++ks) {
            const int k = ks * 4 + kOff;
            v2f a, b;
            a.x = out[rA][k];            a.y = out[rA][k + 1];
            b.x = W1[k * HH + ncol];     b.y = W1[(k + 1) * HH + ncol];
            acc = __builtin_amdgcn_wmma_f32_16x16x4_f32(
                      false, a, false, b, (short)0, acc, false, false);
        }
        const float bias = b1[ncol];
#pragma unroll
        for (int i = 0; i < 8; ++i) y[mB + i][ncol] = acc[i] + bias;
    }

    // LayerNorm + ReLU, one row per lane (lanes 0-15)
    if (lane < 16) {
        float mu = 0.0f;
#pragma unroll 8
        for (int cc = 0; cc < HH; ++cc) mu += y[lane][cc];
        mu *= (1.0f / HH);
        float var = 0.0f;
#pragma unroll 8
        for (int cc = 0; cc < HH; ++cc) { float d0 = y[lane][cc] - mu; var += d0 * d0; }
        const float inv = rsqrtf(var * (1.0f / HH) + LN_EPS);
#pragma unroll 8
        for (int cc = 0; cc < HH; ++cc) {
            float v = (y[lane][cc] - mu) * inv * lg[cc] + lb[cc];
            y[lane][cc] = fmaxf(v, 0.0f);
        }
    }

    // GEMM2: h += y(16x128) @ W2(128x64) + b2   (DeepGCN res+ residual)
    for (int ct = 0; ct < DD / 16; ++ct) {
        v8f acc = {};
        const int ncol = ct * 16 + rA;
#pragma unroll
        for (int ks = 0; ks < HH / 4; ++ks) {
            const int k = ks * 4 + kOff;
            v2f a, b;
            a.x = y[rA][k];              a.y = y[rA][k + 1];
            b.x = W2[k * DD + ncol];     b.y = W2[(k + 1) * DD + ncol];
            acc = __builtin_amdgcn_wmma_f32_16x16x4_f32(
                      false, a, false, b, (short)0, acc, false, false);
        }
        const float bias = b2[ncol];
#pragma unroll
        for (int i = 0; i < 8; ++i) {
            const int node = row0 + mB + i;
            h[node * DD + ncol] += acc[i] + bias;
        }
    }
}

// ---------------------------------------------------------------- launcher
extern "C" void kernel_launch(void* const* d_in, const int* in_sizes, int n_in,
                              void* d_out, int out_size, void* d_ws, size_t ws_size,
                              hipStream_t stream) {
    const float* x   = (const float*)d_in[0];
    const int*   ei  = (const int*)  d_in[1];
    const float* ea  = (const float*)d_in[2];
    const float* nW  = (const float*)d_in[3];
    const float* nb  = (const float*)d_in[4];
    const float* eW  = (const float*)d_in[5];
    const float* eb  = (const float*)d_in[6];
    const float* bng = (const float*)d_in[7];
    const float* bnb = (const float*)d_in[8];
    const float* t   = (const float*)d_in[9];
    const float* W1  = (const float*)d_in[10];
    const float* b1  = (const float*)d_in[11];
    const float* lng = (const float*)d_in[12];
    const float* lnb = (const float*)d_in[13];
    const float* W2  = (const float*)d_in[14];
    const float* b2  = (const float*)d_in[15];
    float* h = (float*)d_out;               // h lives in d_out (same shape)

    float* hn     = (float*)d_ws;
    float* maxbuf = hn     + (size_t)NN * DD;
    float* denom  = maxbuf + (size_t)NN * DD;
    float* num    = denom  + (size_t)NN * DD;
    float* stats  = num    + (size_t)NN * DD;   // 128 floats

    const int gND = (NN * DD + 255) / 256;
    const int gED = (EE * DD + 255) / 256;

    node_enc<<<gND, 256, 0, stream>>>(x, nW, nb, h);

    for (int i = 0; i < LL; ++i) {
        layer_init<<<gND, 256, 0, stream>>>(maxbuf, denom, num, stats);
        bn_stats<<<256, 256, 0, stream>>>(h, stats);
        bn_apply<<<gND, 256, 0, stream>>>(h, stats, bng + i * DD, bnb + i * DD, hn);
        edge_max<<<gED, 256, 0, stream>>>(ei, ea, eW, eb, hn, t + i, maxbuf);
        edge_sum<<<gED, 256, 0, stream>>>(ei, ea, eW, eb, hn, t + i, maxbuf, denom, num);
        mlp_wmma<<<(NT + 3) / 4, 128, 0, stream>>>(
            hn, num, denom,
            W1 + (size_t)i * DD * HH, b1 + (size_t)i * HH,
            lng + (size_t)i * HH, lnb + (size_t)i * HH,
            W2 + (size_t)i * HH * DD, b2 + (size_t)i * DD, h);
    }
}